// OccupancyGridEMABatched_23854248362333
// MI455X (gfx1250) — compile-verified
//
#include <hip/hip_runtime.h>
#include <stdint.h>

// Occupancy-grid EMA scatter-max update for MI455X (gfx1250, wave32).
//
// Semantics (matches JAX reference):
//   touched voxel v : out[v] = max(0.95 * grid[v], max_{points p -> v} occ_val[p])
//   untouched voxel : out[v] = grid[v]            (old, NON-decayed value)
//
// d_out doubles as the scatter buffer (int view, sentinel -1). occ values are
// non-negative floats -> IEEE bits are monotone under signed-int compare, so
// one global_atomic_max_i32 per point implements float scatter-max and the -1
// sentinel sorts below every touched value.
//
// CDNA5-specific path: point stream staged per-tile into LDS with
// GLOBAL_LOAD_ASYNC_TO_LDS_B128 (ASYNCcnt / s_wait_asynccnt). The stream is
// read-once, so it is loaded with th:TH_LOAD_NT to keep the 64MB grid+out
// working set resident in the 192MB L2 for the atomic and finalize phases.

#define THREADS 256
#define TILE    1024     // points per block: pts 12KB + occ 4KB + bidx 4KB = 20KB LDS
#define RES     128
#define EMA     0.95f

// ---------------------------------------------------------------- K1: sentinel
__global__ void occ_k1_init_sentinel(int* __restrict__ out_i, int n4) {
    int t = blockIdx.x * blockDim.x + threadIdx.x;
    if (t < n4) {
        ((int4*)out_i)[t] = make_int4(-1, -1, -1, -1);
    }
}

// -------------------------------------------------------- async copy helpers
__device__ __forceinline__ void async_b128_nt(unsigned lds_addr, uint64_t gaddr) {
    // vdst = LDS byte address (low 32 bits of generic shared pointer),
    // vaddr = 64-bit global address, saddr = off. Tracked by ASYNCcnt.
    // NT: read-once stream, do not retain in cache hierarchy.
    asm volatile("global_load_async_to_lds_b128 %0, %1, off th:TH_LOAD_NT"
                 :: "v"(lds_addr), "v"(gaddr)
                 : "memory");
}

__device__ __forceinline__ void wait_async0() {
    asm volatile("s_wait_asynccnt 0" ::: "memory");
}

// ------------------------------------------------------------- point -> atomic
__device__ __forceinline__ void do_point(float x, float y, float z,
                                         int b, float ov,
                                         int* __restrict__ out_i) {
    // ((p/2 + 0.5) * 128) truncated toward zero, clamped to [0,127]
    float ux = (x * 0.5f + 0.5f) * 128.0f;
    float uy = (y * 0.5f + 0.5f) * 128.0f;
    float uz = (z * 0.5f + 0.5f) * 128.0f;
    int gx = min(max((int)ux, 0), RES - 1);
    int gy = min(max((int)uy, 0), RES - 1);
    int gz = min(max((int)uz, 0), RES - 1);
    int idx = (b << 21) | (gx << 14) | (gy << 7) | gz;
    atomicMax(out_i + idx, __float_as_int(ov));   // no-return global_atomic_max_i32
}

// ---------------------------------------------------------------- K2: scatter
__global__ void occ_k2_scatter(const float* __restrict__ pts,
                               const float* __restrict__ occ,
                               const int*   __restrict__ bidx,
                               int* __restrict__ out_i, int n) {
    __shared__ __align__(16) float s_pts[TILE * 3];  // 12288 B
    __shared__ __align__(16) float s_occ[TILE];      //  4096 B
    __shared__ __align__(16) int   s_bix[TILE];      //  4096 B

    const int t = threadIdx.x;
    const long long base = (long long)blockIdx.x * TILE;

    if (base + TILE <= (long long)n) {
        // Full tile: stage through LDS with async B128 copies.
        // 256 lanes x 16B = 4KB per instruction; pts needs 3, occ 1, bidx 1.
        const unsigned lane_off = (unsigned)(t * 16);

        uint64_t gp = (uint64_t)(uintptr_t)pts + (uint64_t)base * 12u + lane_off;
        unsigned lp = (unsigned)(uintptr_t)(void*)s_pts + lane_off;
        async_b128_nt(lp,         gp);
        async_b128_nt(lp + 4096u, gp + 4096u);
        async_b128_nt(lp + 8192u, gp + 8192u);

        uint64_t go = (uint64_t)(uintptr_t)occ + (uint64_t)base * 4u + lane_off;
        unsigned lo = (unsigned)(uintptr_t)(void*)s_occ + lane_off;
        async_b128_nt(lo, go);

        uint64_t gb = (uint64_t)(uintptr_t)bidx + (uint64_t)base * 4u + lane_off;
        unsigned lb = (unsigned)(uintptr_t)(void*)s_bix + lane_off;
        async_b128_nt(lb, gb);

        wait_async0();       // this wave's 5 async copies done
        __syncthreads();     // all waves' slices visible

        #pragma unroll
        for (int j = 0; j < TILE / THREADS; ++j) {
            int p = t + j * THREADS;   // 12B stride, gcd(3,64)=1 -> no bank conflicts
            do_point(s_pts[p * 3 + 0], s_pts[p * 3 + 1], s_pts[p * 3 + 2],
                     s_bix[p], s_occ[p], out_i);
        }
    } else {
        // Tail tile (not hit for N = 4Mi, but keep it general & uniform per block)
        for (int j = 0; j < TILE / THREADS; ++j) {
            long long p = base + t + j * THREADS;
            if (p < (long long)n) {
                do_point(pts[p * 3 + 0], pts[p * 3 + 1], pts[p * 3 + 2],
                         bidx[p], occ[p], out_i);
            }
        }
    }
}

// --------------------------------------------------------------- K3: finalize
__global__ void occ_k3_finalize(const float* __restrict__ flat,
                                float* __restrict__ out, int n4) {
    int t = blockIdx.x * blockDim.x + threadIdx.x;
    if (t >= n4) return;
    int4   m = ((const int4*)out)[t];     // scatter results / sentinels
    float4 f = ((const float4*)flat)[t];  // original grid
    float4 r;
    r.x = (m.x < 0) ? f.x : fmaxf(EMA * f.x, __int_as_float(m.x));
    r.y = (m.y < 0) ? f.y : fmaxf(EMA * f.y, __int_as_float(m.y));
    r.z = (m.z < 0) ? f.z : fmaxf(EMA * f.z, __int_as_float(m.z));
    r.w = (m.w < 0) ? f.w : fmaxf(EMA * f.w, __int_as_float(m.w));
    ((float4*)out)[t] = r;
}

// ------------------------------------------------------------------- launcher
extern "C" void kernel_launch(void* const* d_in, const int* in_sizes, int n_in,
                              void* d_out, int out_size, void* d_ws, size_t ws_size,
                              hipStream_t stream) {
    (void)n_in; (void)d_ws; (void)ws_size;
    const float* grid = (const float*)d_in[0];   // [B*R^3] f32
    const float* pts  = (const float*)d_in[1];   // [N,3]   f32
    const float* occ  = (const float*)d_in[2];   // [N]     f32
    const int*   bix  = (const int*)  d_in[3];   // [N]     i32
    float* out = (float*)d_out;                  // [B*R^3] f32

    const int nvox = out_size;                   // 8,388,608
    const int npts = in_sizes[2];                // 4,194,304
    const int n4   = nvox / 4;

    const int gridVox = (n4 + THREADS - 1) / THREADS;        // 8192 blocks
    const int nTiles  = (npts + TILE - 1) / TILE;            // 4096 blocks

    occ_k1_init_sentinel<<<gridVox, THREADS, 0, stream>>>((int*)out, n4);
    occ_k2_scatter      <<<nTiles,  THREADS, 0, stream>>>(pts, occ, bix, (int*)out, npts);
    occ_k3_finalize     <<<gridVox, THREADS, 0, stream>>>(grid, out, n4);
}